// NIFTY_19928648253614
// MI455X (gfx1250) — compile-verified
//
#include <hip/hip_runtime.h>

typedef __attribute__((ext_vector_type(2))) float v2f;
typedef __attribute__((ext_vector_type(8))) float v8f;

#define N_FEAT 512
#define N_HID  16

// ---------------------------------------------------------------------------
// deg[i] = 1 (self loop)
__global__ void nifty_init_deg(float* __restrict__ deg, int n) {
    int i = blockIdx.x * blockDim.x + threadIdx.x;
    if (i < n) deg[i] = 1.0f;
}

// deg[col[e]] += 1 for each edge
__global__ void nifty_count_deg(const int* __restrict__ col, int E,
                                float* __restrict__ deg) {
    int e = blockIdx.x * blockDim.x + threadIdx.x;
    if (e < E) atomicAdd(&deg[col[e]], 1.0f);
}

// dinv = rsqrt(deg)  (deg >= 1 always due to self loop)
__global__ void nifty_rsqrt(float* __restrict__ deg, int n) {
    int i = blockIdx.x * blockDim.x + threadIdx.x;
    if (i < n) deg[i] = rsqrtf(deg[i]);
}

// ---------------------------------------------------------------------------
// xw = x @ W via V_WMMA_F32_16X16X4_F32.
// One wave handles a 16-node tile; K=512 -> 128 WMMA steps.
// A (16x4 f32): lanes 0-15 -> M=lane, K={kb,kb+1}; lanes 16-31 -> K={kb+2,kb+3}
// B (4x16 f32): VGPR0 -> K=kb (lo lanes) / kb+2 (hi lanes); VGPR1 -> kb+1 / kb+3
// C/D (16x16 f32): VGPR v -> M=v (lo lanes), M=v+8 (hi lanes); N = lane&15
__global__ void __launch_bounds__(256)
nifty_gemm_wmma(const float* __restrict__ x, const float* __restrict__ W,
                float* __restrict__ xw, int ntiles) {
    __shared__ float Wl[N_FEAT * N_HID];  // 32 KB of the 320 KB WGP LDS
    for (int i = threadIdx.x; i < N_FEAT * N_HID; i += blockDim.x)
        Wl[i] = W[i];
    __syncthreads();

    const int wave = blockIdx.x * (blockDim.x >> 5) + (threadIdx.x >> 5);
    if (wave >= ntiles) return;           // wave-uniform: EXEC stays all-ones
    const int lane = threadIdx.x & 31;
    const int half = lane >> 4;           // 0: lanes 0-15, 1: lanes 16-31
    const int mn   = lane & 15;           // M for A-loads, N for B/C

    const float* xrow = x + (size_t)(wave * 16 + mn) * N_FEAT + 2 * half;
    const float* wcol = Wl + 2 * half * N_HID + mn;

    v8f c = {};
#pragma unroll 8
    for (int kb = 0; kb < N_FEAT; kb += 4) {
        v2f a = *(const v2f*)(xrow + kb);      // global_load_b64, 8B aligned
        v2f b;
        b.x = wcol[kb * N_HID];                // ds_load_b32
        b.y = wcol[kb * N_HID + N_HID];        // ds_load_b32
        c = __builtin_amdgcn_wmma_f32_16x16x4_f32(
                /*neg_a=*/false, a, /*neg_b=*/false, b,
                /*c_mod=*/(short)0, c, /*reuse_a=*/false, /*reuse_b=*/false);
    }

    float* orow = xw + ((size_t)wave * 16 + half * 8) * N_HID + mn;
#pragma unroll
    for (int v = 0; v < 8; ++v)
        orow[v * N_HID] = c[v];
}

// ---------------------------------------------------------------------------
// out = b + dinv^2 * xw   (self-loop term; fully overwrites poisoned d_out)
__global__ void nifty_self_bias(const float* __restrict__ xw,
                                const float* __restrict__ dinv,
                                const float* __restrict__ bias,
                                float* __restrict__ out, int total) {
    int t = blockIdx.x * blockDim.x + threadIdx.x;
    if (t < total) {
        int i = t >> 4, j = t & 15;
        float di = dinv[i];
        out[t] = bias[j] + di * di * xw[t];
    }
}

// out[col] += dinv[row]*dinv[col] * xw[row]; 16 threads per edge.
// xw/dinv/out all L2-resident (192 MB L2), atomics resolve in L2.
__global__ void nifty_scatter(const int* __restrict__ row,
                              const int* __restrict__ col,
                              const float* __restrict__ xw,
                              const float* __restrict__ dinv,
                              float* __restrict__ out, int total) {
    int t = blockIdx.x * blockDim.x + threadIdx.x;
    if (t < total) {
        int e = t >> 4, j = t & 15;
        int r = row[e], cn = col[e];
        float nrm = dinv[r] * dinv[cn];
        atomicAdd(&out[(size_t)cn * N_HID + j], nrm * xw[(size_t)r * N_HID + j]);
    }
}

// ---------------------------------------------------------------------------
extern "C" void kernel_launch(void* const* d_in, const int* in_sizes, int n_in,
                              void* d_out, int out_size, void* d_ws, size_t ws_size,
                              hipStream_t stream) {
    const float* x  = (const float*)d_in[0];
    const int*   ei = (const int*)d_in[1];     // [2, E] flattened
    const float* W  = (const float*)d_in[2];
    const float* b  = (const float*)d_in[3];
    float* out = (float*)d_out;

    const int N = in_sizes[0] / N_FEAT;        // 100000
    const int E = in_sizes[1] / 2;             // 3200000
    const int* row = ei;
    const int* col = ei + E;

    // Workspace: dinv [N] f32, xw [N*16] f32  (~6.8 MB total)
    float* dinv = (float*)d_ws;
    float* xw   = dinv + N;

    nifty_init_deg<<<(N + 255) / 256, 256, 0, stream>>>(dinv, N);
    nifty_count_deg<<<(E + 255) / 256, 256, 0, stream>>>(col, E, dinv);
    nifty_rsqrt<<<(N + 255) / 256, 256, 0, stream>>>(dinv, N);

    const int ntiles = (N + 15) / 16;          // 6250, exact (N % 16 == 0)
    nifty_gemm_wmma<<<(ntiles + 7) / 8, 256, 0, stream>>>(x, W, xw, ntiles);

    const int selfTotal = N * N_HID;
    nifty_self_bias<<<(selfTotal + 255) / 256, 256, 0, stream>>>(xw, dinv, b, out, selfTotal);

    const int edgeTotal = E * N_HID;           // 51.2M
    nifty_scatter<<<(edgeTotal + 255) / 256, 256, 0, stream>>>(row, col, xw, dinv, out, edgeTotal);
}